// MLSWATNeuron_6837587935402
// MI455X (gfx1250) — compile-verified
//
#include <hip/hip_runtime.h>

// MLSWAT spiking-neuron forward scan, MI455X (gfx1250, wave32).
// Memory-bound streaming kernel: 805 MB HBM traffic, ~35 us floor @ 23.3 TB/s.
// No matrix math in this op => no WMMA; optimize the data-movement path:
// b128 non-temporal loads/stores, SGPR-broadcast thresholds, T-scan in
// registers, exact reference FP ordering (threshold select is discontinuous,
// so accumulation order must match the reference bit-for-bit).

typedef float v4f __attribute__((ext_vector_type(4)));

#define T_STEPS 4   // args.T in the reference
#define L_TH    4   // number of thresholds

__device__ __forceinline__ float bcast_sgpr(float x) {
  // Force uniform constants into SGPRs (wave32 broadcast).
  return __uint_as_float(__builtin_amdgcn_readfirstlane(__float_as_uint(x)));
}

// ---------------------------------------------------------------------------
// Prep: th = flip(cumsum(softplus(raw))), aux = zero_point / T.
// ws layout: [0..3] th (descending), [4] aux.
// ---------------------------------------------------------------------------
__global__ void mlswat_prep(const float* __restrict__ raw,
                            const float* __restrict__ zp,
                            float* __restrict__ ws) {
  if (threadIdx.x != 0 || blockIdx.x != 0) return;
  float cum[L_TH];
  float acc = 0.0f;
#pragma unroll
  for (int i = 0; i < L_TH; ++i) {
    float x  = raw[i];
    // numerically stable softplus: max(x,0) + log1p(exp(-|x|))
    float sp = fmaxf(x, 0.0f) + log1pf(expf(-fabsf(x)));
    acc += sp;
    cum[i] = acc;
  }
#pragma unroll
  for (int j = 0; j < L_TH; ++j)
    ws[j] = cum[L_TH - 1 - j];              // flip -> descending
  ws[4] = zp[0] * (1.0f / (float)T_STEPS);  // aux
}

// ---------------------------------------------------------------------------
// Main scan, exact reference order per element per timestep:
//   w = (v + x_t) + aux
//   s = largest th_i <= w, else 0          (first-true over descending th)
//   out_t = s - aux
//   v = w - s                              (soft reset)
// Each thread owns 4 consecutive sites across all T=4 steps; all T loads are
// issued up-front (independent addresses) as non-temporal b128.
// ---------------------------------------------------------------------------
__global__ __launch_bounds__(256) void mlswat_fwd(
    const float* __restrict__ in, float* __restrict__ out,
    const float* __restrict__ ws,
    long long N, long long n4, long long rem) {
  const float th0 = bcast_sgpr(ws[0]);   // largest
  const float th1 = bcast_sgpr(ws[1]);
  const float th2 = bcast_sgpr(ws[2]);
  const float th3 = bcast_sgpr(ws[3]);   // smallest
  const float aux = bcast_sgpr(ws[4]);

  const long long i = (long long)blockIdx.x * blockDim.x + threadIdx.x;

  if (i < n4) {
    // Vector path: N % 4 == 0 guaranteed by launcher.
    const long long N4 = N >> 2;
    const v4f* __restrict__ pin  = (const v4f*)in;
    v4f*       __restrict__ pout = (v4f*)out;

    v4f x[T_STEPS];
#pragma unroll
    for (int t = 0; t < T_STEPS; ++t)
      x[t] = __builtin_nontemporal_load(pin + (long long)t * N4 + i);

    float v[4] = {0.5f, 0.5f, 0.5f, 0.5f};  // INITIAL_MEM
#pragma unroll
    for (int t = 0; t < T_STEPS; ++t) {
      v4f o;
#pragma unroll
      for (int k = 0; k < 4; ++k) {
        float w = (v[k] + x[t][k]) + aux;  // reference association
        float s = 0.0f;
        s = (w >= th3) ? th3 : s;          // ascending overwrite => picks the
        s = (w >= th2) ? th2 : s;          // LARGEST satisfied threshold
        s = (w >= th1) ? th1 : s;
        s = (w >= th0) ? th0 : s;
        o[k] = s - aux;
        v[k] = w - s;
      }
      __builtin_nontemporal_store(o, pout + (long long)t * N4 + i);
    }
  } else if (i < n4 + rem) {
    // Scalar tail / fallback path.
    const long long e = n4 * 4 + (i - n4);
    float v = 0.5f;
#pragma unroll
    for (int t = 0; t < T_STEPS; ++t) {
      float w = (v + in[(long long)t * N + e]) + aux;
      float s = 0.0f;
      s = (w >= th3) ? th3 : s;
      s = (w >= th2) ? th2 : s;
      s = (w >= th1) ? th1 : s;
      s = (w >= th0) ? th0 : s;
      out[(long long)t * N + e] = s - aux;
      v = w - s;
    }
  }
}

extern "C" void kernel_launch(void* const* d_in, const int* in_sizes, int n_in,
                              void* d_out, int out_size, void* d_ws, size_t ws_size,
                              hipStream_t stream) {
  (void)n_in; (void)out_size; (void)ws_size;
  const float* x   = (const float*)d_in[0];   // [T,16,2048,768] f32
  const float* raw = (const float*)d_in[1];   // [4] f32
  const float* zp  = (const float*)d_in[2];   // [1] f32
  float* out = (float*)d_out;
  float* ws  = (float*)d_ws;                  // 5 floats used

  const long long total = (long long)in_sizes[0];
  const long long N     = total / T_STEPS;    // sites per timestep
  long long n4  = N >> 2;
  long long rem = N & 3;
  if (rem) { n4 = 0; rem = N; }               // all-scalar fallback if unaligned

  mlswat_prep<<<dim3(1), dim3(1), 0, stream>>>(raw, zp, ws);

  const int threads = 256;                    // 8 wave32s per workgroup
  long long work = n4 + rem;
  if (work < 1) work = 1;
  const unsigned blocks = (unsigned)((work + threads - 1) / threads);
  mlswat_fwd<<<dim3(blocks), dim3(threads), 0, stream>>>(x, out, ws, N, n4, rem);
}